// GSRNoPooling_32263794328362
// MI455X (gfx1250) — compile-verified
//
#include <hip/hip_runtime.h>
#include <hip/hip_bf16.h>

typedef __attribute__((ext_vector_type(2))) float v2f;
typedef __attribute__((ext_vector_type(8))) float v8f;

#define NEG_SLOPE 0.2f
#define MAXK 36            // max inner dim (layer 1: 36, layers 2-5: 28)

// ---------------------------------------------------------------------------
// Dense projection h = X @ W using V_WMMA_F32_16X16X4_F32.
// One wave computes a 16(row) x 32(col) tile; two 16x16 f32 accumulators.
// W is staged into LDS transposed + zero-padded to 32 columns so all inner-
// loop loads are unconditional 8-byte accesses (global b64 for A, ds b64 for B).
// X: [N,K] row-major, W: [K,C] row-major, H: [N,C] row-major. K % 4 == 0.
// ---------------------------------------------------------------------------
__global__ void gat_linear_wmma(const float* __restrict__ X,
                                const float* __restrict__ W,
                                float* __restrict__ H,
                                int N, int K, int C) {
    __shared__ __align__(16) float Wt[32 * MAXK];   // Wt[col*K + k], cols >= C are 0

    int tid = threadIdx.x;
    for (int i = tid; i < 32 * K; i += blockDim.x) Wt[i] = 0.f;
    __syncthreads();
    for (int i = tid; i < K * C; i += blockDim.x) {
        int k = i / C, c = i % C;
        Wt[c * K + k] = W[i];
    }
    __syncthreads();

    int wave = (int)((blockIdx.x * (unsigned)blockDim.x + tid) >> 5);
    int lane = tid & 31;
    int r0 = wave * 16;
    if (r0 >= N) return;                 // wave-uniform: EXEC all-ones at WMMA

    int half = lane >> 4;                // 0: lanes 0-15, 1: lanes 16-31
    int l16  = lane & 15;
    int rowc = min(r0 + l16, N - 1);     // clamped: OOB rows feed D rows never stored
    int col0 = l16;
    int col1 = 16 + l16;

    const float* xrow = X + (size_t)rowc * K + 2 * half;   // this half's K base
    const float* bt0  = Wt + col0 * K + 2 * half;
    const float* bt1  = Wt + col1 * K + 2 * half;

    v8f acc0 = {};
    v8f acc1 = {};

    for (int k = 0; k < K; k += 4) {
        v2f a  = *(const v2f*)(xrow + k);   // global_load_b64 (8B aligned)
        v2f b0 = *(const v2f*)(bt0 + k);    // ds_load_b64
        v2f b1 = *(const v2f*)(bt1 + k);    // ds_load_b64
        acc0 = __builtin_amdgcn_wmma_f32_16x16x4_f32(false, a, false, b0,
                                                     (short)0, acc0, false, false);
        acc1 = __builtin_amdgcn_wmma_f32_16x16x4_f32(false, a, false, b1,
                                                     (short)0, acc1, false, false);
    }

    // D layout: lane, VGPR v -> row = r0 + v + 8*half, col = l16 (+16 for acc1)
    bool c1ok = col1 < C;
    for (int v = 0; v < 8; ++v) {
        int orow = r0 + v + 8 * half;
        if (orow < N) {
            H[(size_t)orow * C + col0] = acc0[v];
            if (c1ok) H[(size_t)orow * C + col1] = acc1[v];
        }
    }
}

// s[n] = h[n,:]·a_s ; d[n] = h[n,:]·a_d
__global__ void gat_dots(const float* __restrict__ h,
                         const float* __restrict__ a_s,
                         const float* __restrict__ a_d,
                         float* __restrict__ s, float* __restrict__ d,
                         int N, int C) {
    int n = blockIdx.x * blockDim.x + threadIdx.x;
    if (n >= N) return;
    float ss = 0.f, dd = 0.f;
    const float* hr = h + (size_t)n * C;
    for (int c = 0; c < C; ++c) {
        float hv = hr[c];
        ss += hv * a_s[c];
        dd += hv * a_d[c];
    }
    s[n] = ss;
    d[n] = dd;
}

// init per-layer state: m=-inf, denom=0 (i<N); agg=0 (i<N*C)
__global__ void gat_init(float* __restrict__ m, float* __restrict__ denom,
                         float* __restrict__ agg, int N, int NC) {
    int i = blockIdx.x * blockDim.x + threadIdx.x;
    if (i < NC) agg[i] = 0.f;
    if (i < N) { m[i] = -__builtin_inff(); denom[i] = 0.f; }
}

__device__ __forceinline__ void atomicMaxFloat(float* addr, float val) {
    // sign-split trick: positive floats order as ints, negatives reverse as uints
    if (val >= 0.f) atomicMax((int*)addr, __float_as_int(val));
    else            atomicMin((unsigned int*)addr, __float_as_uint(val));
}

__device__ __forceinline__ void edge_endpoints(const int* __restrict__ ei,
                                               int e, int E, int& src, int& tgt) {
    if (e < E) { src = ei[e]; tgt = ei[E + e]; }
    else       { src = tgt = e - E; }          // synthesized self-loop
}

// pass 1: per-target running max of leaky_relu(s[src]+d[tgt])
__global__ void gat_edge_max(const int* __restrict__ ei, int E, int Etot,
                             const float* __restrict__ s, const float* __restrict__ d,
                             float* __restrict__ m) {
    int e = blockIdx.x * blockDim.x + threadIdx.x;
    if (e >= Etot) return;
    int src, tgt; edge_endpoints(ei, e, E, src, tgt);
    float v = s[src] + d[tgt];
    v = v > 0.f ? v : NEG_SLOPE * v;
    atomicMaxFloat(&m[tgt], v);
}

// pass 2: a = exp(e - m[tgt]); denom[tgt] += a; stash a per edge
__global__ void gat_edge_expsum(const int* __restrict__ ei, int E, int Etot,
                                const float* __restrict__ s, const float* __restrict__ d,
                                const float* __restrict__ m,
                                float* __restrict__ a_e, float* __restrict__ denom) {
    int e = blockIdx.x * blockDim.x + threadIdx.x;
    if (e >= Etot) return;
    int src, tgt; edge_endpoints(ei, e, E, src, tgt);
    float v = s[src] + d[tgt];
    v = v > 0.f ? v : NEG_SLOPE * v;
    float a = __expf(v - m[tgt]);
    a_e[e] = a;
    atomicAdd(&denom[tgt], a);
}

// pass 3: agg[tgt,:] += (a_e/denom[tgt]) * h[src,:]; one lane per feature,
// so the 28 atomics per edge hit consecutive addresses (few L2 lines).
__global__ void gat_edge_aggregate(const int* __restrict__ ei, int E, int Etot,
                                   const float* __restrict__ h,
                                   const float* __restrict__ a_e,
                                   const float* __restrict__ denom,
                                   float* __restrict__ agg, int C) {
    long long gt = (long long)blockIdx.x * blockDim.x + threadIdx.x;
    int e    = (int)(gt >> 5);
    int lane = (int)(gt & 31);
    if (e >= Etot || lane >= C) return;
    int src, tgt; edge_endpoints(ei, e, E, src, tgt);
    float coef = a_e[e] / denom[tgt];
    atomicAdd(&agg[(size_t)tgt * C + lane], coef * h[(size_t)src * C + lane]);
}

// h_next = tanh(agg + b)   (in-place on agg)
__global__ void gat_finalize(float* __restrict__ agg, const float* __restrict__ b,
                             int NC, int C) {
    int i = blockIdx.x * blockDim.x + threadIdx.x;
    if (i >= NC) return;
    agg[i] = tanhf(agg[i] + b[i % C]);
}

// node_preds[n] = h[n,:]·Wout + bout
__global__ void gat_node_pred(const float* __restrict__ h,
                              const float* __restrict__ Wout,
                              const float* __restrict__ bout,
                              float* __restrict__ out, int N, int C) {
    int n = blockIdx.x * blockDim.x + threadIdx.x;
    if (n >= N) return;
    float p = bout[0];
    const float* hr = h + (size_t)n * C;
    for (int c = 0; c < C; ++c) p += hr[c] * Wout[c];
    out[n] = p;
}

__global__ void gat_last_init(int* __restrict__ last, int B) {
    int g = blockIdx.x * blockDim.x + threadIdx.x;
    if (g < B) last[g] = 0;
}

__global__ void gat_last_scan(const int* __restrict__ batch, int Nb,
                              int* __restrict__ last) {
    int i = blockIdx.x * blockDim.x + threadIdx.x;
    if (i >= Nb) return;
    atomicMax(&last[batch[i]], i);
}

__global__ void gat_graph_gather(const float* __restrict__ node_preds,
                                 const int* __restrict__ last,
                                 float* __restrict__ graph_preds, int B) {
    int g = blockIdx.x * blockDim.x + threadIdx.x;
    if (g >= B) return;
    graph_preds[g] = node_preds[last[g]];
}

static inline int cdiv(long long a, long long b) { return (int)((a + b - 1) / b); }

extern "C" void kernel_launch(void* const* d_in, const int* in_sizes, int n_in,
                              void* d_out, int out_size, void* d_ws, size_t ws_size,
                              hipStream_t stream) {
    const float* x       = (const float*)d_in[0];
    const int*   ei      = (const int*)d_in[1];
    const int*   batch   = (const int*)d_in[2];
    const float* W1      = (const float*)d_in[4];
    const float* a_s1    = (const float*)d_in[5];
    const float* a_d1    = (const float*)d_in[6];
    const float* b1      = (const float*)d_in[7];
    const float* Ws      = (const float*)d_in[8];
    const float* a_ss    = (const float*)d_in[9];
    const float* a_ds    = (const float*)d_in[10];
    const float* bs      = (const float*)d_in[11];
    const float* Wout    = (const float*)d_in[12];
    const float* bout    = (const float*)d_in[13];

    const int C   = in_sizes[5];              // 28
    const int IN  = in_sizes[4] / C;          // 36
    const int N   = in_sizes[0] / IN;         // 100000
    const int E   = in_sizes[1] / 2;          // 3.2M
    const int Nb  = in_sizes[2];              // batch array length
    const int B   = out_size - N;             // num_graphs
    const int Etot = E + N;                   // + self loops
    const int NC  = N * C;

    // workspace carve-out
    float* hA    = (float*)d_ws;              // [N*C] linear output h
    float* hB    = hA + NC;                   // [N*C] agg / layer input
    float* sbuf  = hB + NC;                   // [N]
    float* dbuf  = sbuf + N;                  // [N]
    float* mbuf  = dbuf + N;                  // [N]
    float* denom = mbuf + N;                  // [N]
    float* a_e   = denom + N;                 // [Etot]
    int*   last  = (int*)(a_e + Etot);        // [B]

    const int TB = 256;
    const int gemm_blocks = cdiv((long long)cdiv(N, 16) * 32, TB);
    const int node_blocks = cdiv(N, TB);
    const int nc_blocks   = cdiv(NC, TB);
    const int edge_blocks = cdiv(Etot, TB);
    const int aggr_blocks = cdiv((long long)Etot * 32, TB);

    float* out_nodes  = (float*)d_out;        // [N]
    float* out_graphs = out_nodes + N;        // [B]

    for (int l = 0; l < 5; ++l) {
        const float* Xin = (l == 0) ? x   : hB;
        const int    K   = (l == 0) ? IN  : C;
        const float* W   = (l == 0) ? W1  : (Ws   + (size_t)(l - 1) * C * C);
        const float* as_ = (l == 0) ? a_s1 : (a_ss + (size_t)(l - 1) * C);
        const float* ad_ = (l == 0) ? a_d1 : (a_ds + (size_t)(l - 1) * C);
        const float* bb  = (l == 0) ? b1  : (bs   + (size_t)(l - 1) * C);

        gat_linear_wmma<<<gemm_blocks, TB, 0, stream>>>(Xin, W, hA, N, K, C);
        gat_dots<<<node_blocks, TB, 0, stream>>>(hA, as_, ad_, sbuf, dbuf, N, C);
        gat_init<<<nc_blocks, TB, 0, stream>>>(mbuf, denom, hB, N, NC);
        gat_edge_max<<<edge_blocks, TB, 0, stream>>>(ei, E, Etot, sbuf, dbuf, mbuf);
        gat_edge_expsum<<<edge_blocks, TB, 0, stream>>>(ei, E, Etot, sbuf, dbuf,
                                                        mbuf, a_e, denom);
        gat_edge_aggregate<<<aggr_blocks, TB, 0, stream>>>(ei, E, Etot, hA, a_e,
                                                           denom, hB, C);
        gat_finalize<<<nc_blocks, TB, 0, stream>>>(hB, bb, NC, C);
    }

    gat_node_pred<<<node_blocks, TB, 0, stream>>>(hB, Wout, bout, out_nodes, N, C);
    gat_last_init<<<cdiv(B, TB), TB, 0, stream>>>(last, B);
    gat_last_scan<<<cdiv(Nb, TB), TB, 0, stream>>>(batch, Nb, last);
    gat_graph_gather<<<cdiv(B, TB), TB, 0, stream>>>(out_nodes, last, out_graphs, B);
}